// AngularLSHTriton_51994874085513
// MI455X (gfx1250) — compile-verified
//
#include <hip/hip_runtime.h>
#include <hip/hip_bf16.h>

// Angular LSH bucketization for MI455X (gfx1250, wave32, WMMA).
//
// scores[b,h,n,r] = mat[b,h,n,:] . proj[:,r];  bits = pack(sign(scores));
// out = perm[bits]
//
// Roofline: 134 MB streamed in / 2 MB out @ 23.3 TB/s ~= 5.9 us floor;
// 0.54 GFLOP -> needs only ~92 TF fp32 of matrix math, so V_WMMA_F32_16X16X4_F32
// (full fp32, preserves sign fidelity near zero) is nowhere near the bottleneck.
// Design goals: minimize vmem *requests* per tile, keep b128-wide streaming,
// amortize per-wave setup over 4 consecutive tiles, double-buffer LDS so the
// next tile's global loads overlap the current tile's WMMA chain.

typedef __attribute__((ext_vector_type(2))) float v2f;
typedef __attribute__((ext_vector_type(4))) float v4f;
typedef __attribute__((ext_vector_type(8))) float v8f;

#define HEADDIM 64
#define NUM_PROJS 8
#define ROWS_PER_TILE 16
#define WAVES_PER_BLOCK 8
#define TILES_PER_WAVE 4
#define TILE_DWORDS (ROWS_PER_TILE * HEADDIM)   // 1024 dwords = 4 KB
#define LDS_STRIDE 68   // 64 + 4 pad dwords: float4-aligned, bank-conflict-free frag reads

__global__ __launch_bounds__(256)
void angular_lsh_wmma_kernel(const float* __restrict__ mat,
                             const float* __restrict__ proj,   // [64][8] row-major
                             const int* __restrict__ perm,     // [256]
                             int* __restrict__ out,
                             int nTiles) {
    // Double-buffered per-wave staging: 8 waves x 2 x (16*68*4B) = 68 KB/WG.
    __shared__ float lds[WAVES_PER_BLOCK][2][ROWS_PER_TILE * LDS_STRIDE];

    const int lane  = threadIdx.x & 31;
    const int wave  = threadIdx.x >> 5;
    const int tile0 = (blockIdx.x * WAVES_PER_BLOCK + wave) * TILES_PER_WAVE; // wave-uniform
    if (tile0 >= nTiles) return;   // uniform exit: EXEC stays all-1 for WMMA

    const int row16 = lane & 15;
    const int halfw = lane >> 4;

    // ---- Build B fragments ONCE per wave (projection, zero-padded cols 8..15) ----
    // ISA layout for 4x16 fp32 B: VGPR0 = K=0 (lanes 0-15) / K=2 (lanes 16-31),
    // VGPR1 = K=1 / K=3.  For K-step s, lane supplies B[4s+2*half + {0,1}][lane&15].
    v2f bfrag[16];
#pragma unroll
    for (int s = 0; s < 16; ++s) {
        const int k0 = 4 * s + 2 * halfw;
        float b0 = 0.0f, b1 = 0.0f;
        if (row16 < NUM_PROJS) {
            b0 = proj[k0 * NUM_PROJS + row16];
            b1 = proj[(k0 + 1) * NUM_PROJS + row16];
        }
        bfrag[s] = (v2f){b0, b1};
    }

    // ---- Prologue: fetch tile0 (contiguous 4 KB) and stage into LDS buf 0 ----
    // Streamed-once data: non-temporal loads keep it from rinsing L2 at RT priority.
    v4f pf[8];
    {
        const float* src = mat + (size_t)tile0 * TILE_DWORDS;
#pragma unroll
        for (int i = 0; i < 8; ++i)
            pf[i] = __builtin_nontemporal_load((const v4f*)(src + i * 128 + lane * 4));
        float* buf = &lds[wave][0][0];
#pragma unroll
        for (int i = 0; i < 8; ++i) {
            const int off = i * 128 + lane * 4;
            *(v4f*)(buf + (off >> 6) * LDS_STRIDE + (off & 63)) = pf[i];
        }
    }

#pragma unroll
    for (int t = 0; t < TILES_PER_WAVE; ++t) {
        const int tile = tile0 + t;
        if (tile >= nTiles) break;                       // uniform
        const bool hasNext = (t + 1 < TILES_PER_WAVE) && (tile + 1 < nTiles);

        // ---- Prefetch next tile into registers (overlaps the WMMA chain) ----
        if (hasNext) {
            const float* nsrc = mat + (size_t)(tile + 1) * TILE_DWORDS;
#pragma unroll
            for (int i = 0; i < 8; ++i)
                pf[i] = __builtin_nontemporal_load((const v4f*)(nsrc + i * 128 + lane * 4));
        }

        // ---- 16 chained V_WMMA_F32_16X16X4_F32: C(16x16) += A(16x4) x B(4x16) ----
        // A fragment per step s: lane holds {A[m][4s+2*half], A[m][4s+2*half+1]}, m=lane&15.
        // LDS_STRIDE=68 makes these ds_load_b64 reads bank-conflict-free.
        const v2f* arow = (const v2f*)(&lds[wave][t & 1][0] + row16 * LDS_STRIDE);
        v8f c = {0.f, 0.f, 0.f, 0.f, 0.f, 0.f, 0.f, 0.f};
#pragma unroll
        for (int s = 0; s < 16; ++s) {
            v2f a = arow[2 * s + halfw];
            c = __builtin_amdgcn_wmma_f32_16x16x4_f32(
                    /*neg_a=*/false, a, /*neg_b=*/false, bfrag[s],
                    /*c_mod=*/(short)0, c, /*reuse_a=*/false, /*reuse_b=*/false);
        }

        // ---- Sign-bit packing via wave32 ballot ----
        // C layout: VGPR v, lanes 0-7  -> score[row v    ][proj 0..7]
        //                   lanes16-23 -> score[row v + 8][proj 0..7]
        // ballot bit r carries sign of proj r -> exactly the 2^r encoding (enc_vec).
        unsigned mmask[8];
#pragma unroll
        for (int v = 0; v < 8; ++v)
            mmask[v] = (unsigned)__ballot(c[v] > 0.0f);

        unsigned sel = mmask[0];
#pragma unroll
        for (int v = 1; v < 8; ++v)
            sel = ((lane & 7) == v) ? mmask[v] : sel;

        if (lane < 16) {   // divergent, but after the WMMA chain; EXEC restored below
            const unsigned bits = (sel >> ((lane >> 3) * 16)) & 0xffu;
            out[(size_t)tile * ROWS_PER_TILE + lane] = perm[bits];
        }

        // ---- Stage prefetched tile into the other LDS buffer ----
        if (hasNext) {
            float* nbuf = &lds[wave][(t + 1) & 1][0];
#pragma unroll
            for (int i = 0; i < 8; ++i) {
                const int off = i * 128 + lane * 4;
                *(v4f*)(nbuf + (off >> 6) * LDS_STRIDE + (off & 63)) = pf[i];
            }
        }
    }
}

extern "C" void kernel_launch(void* const* d_in, const int* in_sizes, int n_in,
                              void* d_out, int out_size, void* d_ws, size_t ws_size,
                              hipStream_t stream) {
    const float* mat  = (const float*)d_in[0];   // [2,32,8192,64] fp32
    const float* proj = (const float*)d_in[1];   // [1,1,64,8]     fp32
    const int*   perm = (const int*)d_in[2];     // [256]          int32
    // d_in[3] (enc_vec) is the fixed 2^r weighting -- realized by ballot bit order.
    int* out = (int*)d_out;                      // int32 buckets, one per row

    const int nRows  = in_sizes[0] / HEADDIM;            // 524,288
    const int nTiles = nRows / ROWS_PER_TILE;            // 32,768

    const int tilesPerBlock = WAVES_PER_BLOCK * TILES_PER_WAVE;   // 32
    dim3 block(32 * WAVES_PER_BLOCK);
    dim3 grid((nTiles + tilesPerBlock - 1) / tilesPerBlock);      // 1024 blocks
    angular_lsh_wmma_kernel<<<grid, block, 0, stream>>>(mat, proj, perm, out, nTiles);
}